// Attention_38276748542551
// MI455X (gfx1250) — compile-verified
//
#include <hip/hip_runtime.h>
#include <hip/hip_bf16.h>
#include <cstdint>
#include <cstddef>

// ---------------------------------------------------------------------------
// Problem constants (from the reference)
// ---------------------------------------------------------------------------
#define DIMX      4096
#define N_HEADS   32
#define N_KV      8
#define HEAD_DIM  128
#define MAX_BS    2
#define MAX_SEQ   2048
#define BATCH     2
#define SEQ       2048
#define NTOK      (BATCH * SEQ)          // 4096 tokens
#define KV_DIM    (N_KV * HEAD_DIM)      // 1024

typedef float v2f __attribute__((ext_vector_type(2)));
typedef float v8f __attribute__((ext_vector_type(8)));

// Full-precision fp32 matrix op: D = A(16x4) * B(4x16) + C.
static __device__ __forceinline__ v8f wmma4(v2f a, v2f b, v8f c) {
    return __builtin_amdgcn_wmma_f32_16x16x4_f32(
        /*neg_a=*/false, a, /*neg_b=*/false, b,
        /*c_mod=*/(short)0, c, /*reuse_a=*/false, /*reuse_b=*/false);
}

// ---------------------------------------------------------------------------
// CDNA5 async global->LDS copy (ASYNCcnt-tracked, bypasses VGPRs).
// LDS destination operand is the workgroup-relative byte offset; per ISA
// section 10.2 the low 32 bits of a flat shared-aperture address ARE that
// offset, so truncating the generic pointer is exact.
// ---------------------------------------------------------------------------
static __device__ __forceinline__ unsigned lds_off(const void* p) {
    return (unsigned)(size_t)p;
}
static __device__ __forceinline__ void async_ld_b128(unsigned dst_lds, const float* g) {
    asm volatile("global_load_async_to_lds_b128 %0, %1, off"
                 :: "v"(dst_lds), "v"(g) : "memory");
}
template <int N>
static __device__ __forceinline__ void wait_async() {
    asm volatile("s_wait_asynccnt %0" :: "n"(N) : "memory");
}

// ---------------------------------------------------------------------------
// GEMM: C[M,N] = A[M,K] @ B[N,K]^T   (fp32, WMMA 16x16x4)
// Block tile 128x128, K-step 32, 8 waves: wave tile 32(M) x 64(N) = 2x4 accs.
// Double-buffered LDS panels filled by async-to-LDS copies.
// mode 0: C row-major (ldc = N)
// mode 1: scatter into KV panel [b][kv][start_pos + t][HEAD_DIM]
// ---------------------------------------------------------------------------
#define LSTR  36                 // row stride (floats): 16B-aligned rows, 36m%64 distinct
#define PANEL (128 * LSTR)       // one A or B panel, in floats
#define GEMM_LDS_BYTES (4 * PANEL * 4)   // A0,B0,A1,B1 = 73,728 B (CDNA5: 320KB WGP LDS)

__global__ __launch_bounds__(256)
void gemm_nt_wmma(const float* __restrict__ A, const float* __restrict__ Bm,
                  float* __restrict__ C, int M, int N, int K,
                  int mode, const int* __restrict__ start_pos)
{
    extern __shared__ float gsm[];

    const int tid  = threadIdx.x;
    const int lane = tid & 31;
    const int w    = tid >> 5;
    const int wm   = w & 3;        // 4 wave-rows  -> 32 M rows each
    const int wn   = w >> 2;       // 2 wave-cols  -> 64 N cols each
    const int hi   = lane >> 4;    // half-wave: K pair selector / M+8
    const int l15  = lane & 15;

    const size_t mBase = (size_t)blockIdx.y * 128;
    const size_t nBase = (size_t)blockIdx.x * 128;

    v8f acc[2][4];
    for (int im = 0; im < 2; ++im)
        for (int in = 0; in < 4; ++in)
            acc[im][in] = (v8f){0.f,0.f,0.f,0.f,0.f,0.f,0.f,0.f};

    const int ldr = tid >> 3;          // 0..31  (row within 32-row chunk)
    const int ldc = (tid & 7) * 4;     // 0..28  (float4 column)
    const unsigned ldsBase = lds_off(gsm);

    // Issue one 128x32 A panel + B panel into buffer `buf` (8 async b128/thread).
    auto stage = [&](int buf, int k0) {
        const unsigned offA = ldsBase + (unsigned)(buf * 2 * PANEL) * 4u;
        const unsigned offB = offA + (unsigned)PANEL * 4u;
        for (int i = 0; i < 4; ++i) {
            const int row = ldr + i * 32;
            async_ld_b128(offA + (unsigned)(row * LSTR + ldc) * 4u,
                          A + (mBase + row) * (size_t)K + k0 + ldc);
            async_ld_b128(offB + (unsigned)(row * LSTR + ldc) * 4u,
                          Bm + (nBase + row) * (size_t)K + k0 + ldc);
        }
    };

    const int nk = K >> 5;
    stage(0, 0);

    for (int kt = 0; kt < nk; ++kt) {
        // Prefetch next panels into the other buffer while waiting on this one.
        if (kt + 1 < nk) {
            stage((kt + 1) & 1, (kt + 1) << 5);
            wait_async<8>();            // this stage's 8 copies are older -> done
        } else {
            wait_async<0>();
        }
        __syncthreads();                // all waves' panel data visible

        const float* sA = gsm + (size_t)(kt & 1) * 2 * PANEL;
        const float* sB = sA + PANEL;

        for (int kk = 0; kk < 32; kk += 4) {
            const int kof = kk + 2 * hi;        // lanes 0-15: k,k+1 ; 16-31: k+2,k+3
            v2f a[2], b[4];
            for (int im = 0; im < 2; ++im)
                a[im] = *(const v2f*)&sA[(wm * 32 + im * 16 + l15) * LSTR + kof];
            for (int in = 0; in < 4; ++in)
                b[in] = *(const v2f*)&sB[(wn * 64 + in * 16 + l15) * LSTR + kof];
            for (int im = 0; im < 2; ++im)
                for (int in = 0; in < 4; ++in)
                    acc[im][in] = wmma4(a[im], b[in], acc[im][in]);
        }
        __syncthreads();                // done reading before buffer is re-filled
    }

    // Epilogue.  C-layout: VGPR r -> row (r + 8*hi), col l15 within 16x16 tile.
    for (int im = 0; im < 2; ++im)
        for (int in = 0; in < 4; ++in)
            for (int r = 0; r < 8; ++r) {
                const int mg = (int)mBase + wm * 32 + im * 16 + r + 8 * hi;
                const int ng = (int)nBase + wn * 64 + in * 16 + l15;
                const float v = acc[im][in][r];
                if (mode == 0) {
                    C[(size_t)mg * N + ng] = v;
                } else {
                    const int b  = mg / SEQ, t = mg % SEQ;
                    const int kv = ng >> 7,  d = ng & 127;
                    const int pos = start_pos[0] + t;
                    if (pos < MAX_SEQ)
                        C[(((size_t)(b * N_KV + kv)) * MAX_SEQ + pos) * HEAD_DIM + d] = v;
                }
            }
}

// ---------------------------------------------------------------------------
// RoPE on Q (in place) and K (scattered into panel [b][kv][pos][d]).
// ---------------------------------------------------------------------------
__global__ __launch_bounds__(256)
void rope_scatter(float* __restrict__ q, const float* __restrict__ kx,
                  float* __restrict__ kpanel,
                  const float* __restrict__ cosb, const float* __restrict__ sinb,
                  const int* __restrict__ start_pos)
{
    const int idx = blockIdx.x * 256 + threadIdx.x;   // NTOK * N_HEADS * 64
    const int p  = idx & 63;
    const int h  = (idx >> 6) & 31;
    const int tg = idx >> 11;              // global token 0..NTOK-1
    const int t  = tg & (SEQ - 1);
    const int b  = tg >> 11;               // SEQ == 2048 == 2^11

    const float c  = cosb[t * 64 + p];
    const float sn = sinb[t * 64 + p];

    float* qp = q + (size_t)tg * (N_HEADS * HEAD_DIM) + h * HEAD_DIM + 2 * p;
    const float q0 = qp[0], q1 = qp[1];
    qp[0] = q0 * c - q1 * sn;
    qp[1] = q0 * sn + q1 * c;

    if (h < N_KV) {
        const float* kp = kx + (size_t)tg * KV_DIM + h * HEAD_DIM + 2 * p;
        const float k0 = kp[0], k1 = kp[1];
        const int pos = start_pos[0] + t;
        if (pos < MAX_SEQ) {
            float* dst = kpanel + (((size_t)(b * N_KV + h)) * MAX_SEQ + pos) * HEAD_DIM + 2 * p;
            dst[0] = k0 * c - k1 * sn;
            dst[1] = k0 * sn + k1 * c;
        }
    }
}

// ---------------------------------------------------------------------------
// Copy pre-existing cache entries (t < start_pos) into the K/V panels.
// Never mutates the input caches.  No-op when start_pos == 0.
// ---------------------------------------------------------------------------
__global__ __launch_bounds__(256)
void cache_prefix(const float* __restrict__ ck, const float* __restrict__ cv,
                  float* __restrict__ kpanel, float* __restrict__ vpanel,
                  const int* __restrict__ start_pos)
{
    const int sp  = start_pos[0];
    const int idx = blockIdx.x * 256 + threadIdx.x;   // MAX_BS*MAX_SEQ*N_KV*HEAD_DIM
    const int d   = idx & 127;
    const int kv  = (idx >> 7) & 7;
    const int t   = (idx >> 10) & (MAX_SEQ - 1);
    const int b   = idx >> 21;
    if (t < sp) {
        const size_t src = (((size_t)(b * MAX_SEQ + t)) * N_KV + kv) * HEAD_DIM + d;
        const size_t dst = (((size_t)(b * N_KV + kv)) * MAX_SEQ + t) * HEAD_DIM + d;
        kpanel[dst] = ck[src];
        vpanel[dst] = cv[src];
    }
}

// ---------------------------------------------------------------------------
// Attention core: one block = (batch, head, 16-row query tile).
// LDS: Q tile (16x128), 128-token K/V stage (async-filled), 16 x kv score panel.
// Phase 1: S = Q K^T via wmma; scale + causal mask in C-layout registers.
// Phase 2: row softmax (2 rows / wave, shfl_xor wave32 reductions).
// Phase 3: O = P V via wmma (probabilities read from LDS as A-fragments).
// Out-of-range tokens are never staged: their scores are masked to -1e30 and
// their probabilities are exactly 0, so stale (finite) LDS rows are harmless.
// ---------------------------------------------------------------------------
#define QSTR 132    // 4m%64 distinct for m<16 -> conflict-free b64 fragment reads
#define KSTR 132    // also 16B-aligned rows (132*4 % 16 == 0) for async b128
#define SSTR 2052
#define ATTN_LDS_BYTES ((16 * QSTR + 128 * KSTR + 16 * SSTR) * 4)   // 207,360 B

__global__ __launch_bounds__(256, 1)
void attn_wmma(const float* __restrict__ q, const float* __restrict__ kpanel,
               const float* __restrict__ vpanel, float* __restrict__ attnout,
               const int* __restrict__ start_pos)
{
    extern __shared__ float smem[];
    float* sQ  = smem;                    // 16  * QSTR
    float* sKV = sQ  + 16  * QSTR;        // 128 * KSTR
    float* sS  = sKV + 128 * KSTR;        // 16  * SSTR

    const int tid  = threadIdx.x;
    const int lane = tid & 31;
    const int w    = tid >> 5;
    const int hi   = lane >> 4;
    const int l15  = lane & 15;

    const int qt = blockIdx.x & 127;              // query tile (16 rows)
    const int h  = (blockIdx.x >> 7) & 31;        // head
    const int b  = blockIdx.x >> 12;              // batch
    const int g  = h >> 2;                        // kv head (N_REP = 4)

    const int sp = start_pos[0];
    int kv_len = sp + SEQ;
    if (kv_len > MAX_SEQ) kv_len = MAX_SEQ;
    const int nkt  = (kv_len + 127) >> 7;
    const int plen = nkt << 7;
    const float scale = 0.088388347648318447f;    // 1/sqrt(128)

    // ---- load Q tile (16 x 128) ----
    {
        const int row = tid >> 4;
        const int c0  = (tid & 15) * 8;
        const float* src = q + ((size_t)(b * SEQ + qt * 16 + row)) * (N_HEADS * HEAD_DIM)
                             + h * HEAD_DIM + c0;
        for (int j = 0; j < 8; ++j) sQ[row * QSTR + c0 + j] = src[j];
    }

    const float* Kp = kpanel + ((size_t)(b * N_KV + g)) * MAX_SEQ * HEAD_DIM;
    const float* Vp = vpanel + ((size_t)(b * N_KV + g)) * MAX_SEQ * HEAD_DIM;
    const unsigned kvBase = lds_off(sKV);

    // Async-stage a 128-token x 128-dim tile (16 b128 copies per thread).
    auto stage_tile = [&](const float* Src, int base) {
        for (int i = 0; i < 16; ++i) {
            const int lin = i * 256 + tid;
            const int row = lin >> 5;
            const int c4  = (lin & 31) * 4;
            const int tok = base + row;
            if (tok < kv_len)
                async_ld_b128(kvBase + (unsigned)(row * KSTR + c4) * 4u,
                              Src + (size_t)tok * HEAD_DIM + c4);
        }
    };

    // ---- Phase 1: scores ----
    for (int kt = 0; kt < nkt; ++kt) {
        __syncthreads();                       // prior tile fully consumed
        stage_tile(Kp, kt * 128);
        wait_async<0>();
        __syncthreads();

        const int n0 = w * 16;                 // 8 waves x 16 tokens
        v8f acc = (v8f){0.f,0.f,0.f,0.f,0.f,0.f,0.f,0.f};
        for (int kk = 0; kk < HEAD_DIM; kk += 4) {
            const int kof = kk + 2 * hi;
            const v2f a  = *(const v2f*)&sQ[l15 * QSTR + kof];
            const v2f bb = *(const v2f*)&sKV[(n0 + l15) * KSTR + kof];
            acc = wmma4(a, bb, acc);
        }
        for (int r = 0; r < 8; ++r) {
            const int m    = r + 8 * hi;
            const int qpos = sp + qt * 16 + m;
            const int kpos = kt * 128 + n0 + l15;
            float v = acc[r] * scale;
            if (kpos > qpos || kpos >= kv_len) v = -1e30f;
            sS[m * SSTR + kt * 128 + n0 + l15] = v;
        }
    }
    __syncthreads();

    // ---- Phase 2: softmax over padded length (padding holds -1e30 -> exp 0) ----
    for (int rr = 0; rr < 2; ++rr) {
        const int row = w * 2 + rr;
        float* Sr = sS + row * SSTR;
        float mx = -3.0e38f;
        for (int j = lane; j < plen; j += 32) mx = fmaxf(mx, Sr[j]);
        for (int o = 16; o > 0; o >>= 1) mx = fmaxf(mx, __shfl_xor(mx, o, 32));
        float sum = 0.f;
        for (int j = lane; j < plen; j += 32) {
            const float e = __expf(Sr[j] - mx);
            Sr[j] = e;
            sum += e;
        }
        for (int o = 16; o > 0; o >>= 1) sum += __shfl_xor(sum, o, 32);
        const float inv = 1.0f / sum;
        for (int j = lane; j < plen; j += 32) Sr[j] *= inv;
    }

    // ---- Phase 3: O = P @ V ----
    const int n0 = w * 16;                     // 8 waves x 16 head-dims
    v8f acc = (v8f){0.f,0.f,0.f,0.f,0.f,0.f,0.f,0.f};
    for (int vt = 0; vt < nkt; ++vt) {
        __syncthreads();
        stage_tile(Vp, vt * 128);
        wait_async<0>();
        __syncthreads();

        for (int kk = 0; kk < 128; kk += 4) {
            const int kof = kk + 2 * hi;
            const v2f a = *(const v2f*)&sS[l15 * SSTR + vt * 128 + kof];
            v2f bb;                            // B[k][n] = V[tok k][dim n]
            bb.x = sKV[(kof + 0) * KSTR + n0 + l15];
            bb.y = sKV[(kof + 1) * KSTR + n0 + l15];
            acc = wmma4(a, bb, acc);
        }
    }
    for (int r = 0; r < 8; ++r) {
        const int m = r + 8 * hi;
        attnout[((size_t)(b * SEQ + qt * 16 + m)) * (N_HEADS * HEAD_DIM)
                + h * HEAD_DIM + n0 + l15] = acc[r];
    }
}

// ---------------------------------------------------------------------------
// Launch.  Workspace layout (floats):
//   q       : NTOK*DIM                     (67.1 MB)
//   kx      : NTOK*KV_DIM                  (16.8 MB)
//   kpanel  : MAX_BS*N_KV*MAX_SEQ*HEAD_DIM (16.8 MB)
//   vpanel  : same                         (16.8 MB)
//   attn    : NTOK*DIM                     (67.1 MB)   total ~185 MB
// ---------------------------------------------------------------------------
extern "C" void kernel_launch(void* const* d_in, const int* in_sizes, int n_in,
                              void* d_out, int out_size, void* d_ws, size_t ws_size,
                              hipStream_t stream)
{
    (void)in_sizes; (void)n_in; (void)out_size; (void)ws_size;
    const float* x  = (const float*)d_in[0];
    const float* wq = (const float*)d_in[1];
    const float* wk = (const float*)d_in[2];
    const float* wv = (const float*)d_in[3];
    const float* wo = (const float*)d_in[4];
    const float* fc = (const float*)d_in[5];
    const float* fs = (const float*)d_in[6];
    const float* ck = (const float*)d_in[7];
    const float* cv = (const float*)d_in[8];
    const int*   sp = (const int*)d_in[9];
    float* out = (float*)d_out;

    float* qb     = (float*)d_ws;
    float* kx     = qb + (size_t)NTOK * DIMX;
    float* kpanel = kx + (size_t)NTOK * KV_DIM;
    float* vpanel = kpanel + (size_t)MAX_BS * N_KV * MAX_SEQ * HEAD_DIM;
    float* attn   = vpanel + (size_t)MAX_BS * N_KV * MAX_SEQ * HEAD_DIM;

    const dim3 blk(256);

    (void)hipFuncSetAttribute((const void*)gemm_nt_wmma,
                              hipFuncAttributeMaxDynamicSharedMemorySize,
                              GEMM_LDS_BYTES);
    (void)hipFuncSetAttribute((const void*)attn_wmma,
                              hipFuncAttributeMaxDynamicSharedMemorySize,
                              ATTN_LDS_BYTES);

    // QKV projections (V scattered straight into its panel).
    gemm_nt_wmma<<<dim3(DIMX / 128, NTOK / 128), blk, GEMM_LDS_BYTES, stream>>>(
        x, wq, qb, NTOK, DIMX, DIMX, 0, sp);
    gemm_nt_wmma<<<dim3(KV_DIM / 128, NTOK / 128), blk, GEMM_LDS_BYTES, stream>>>(
        x, wk, kx, NTOK, KV_DIM, DIMX, 0, sp);
    gemm_nt_wmma<<<dim3(KV_DIM / 128, NTOK / 128), blk, GEMM_LDS_BYTES, stream>>>(
        x, wv, vpanel, NTOK, KV_DIM, DIMX, 1, sp);

    // Existing cache prefix (no-op at start_pos==0) + RoPE/scatter.
    cache_prefix<<<(MAX_BS * MAX_SEQ * N_KV * HEAD_DIM) / 256, blk, 0, stream>>>(
        ck, cv, kpanel, vpanel, sp);
    rope_scatter<<<(NTOK * N_HEADS * 64) / 256, blk, 0, stream>>>(
        qb, kx, kpanel, fc, fs, sp);

    // Attention core (207,360 B dynamic LDS — CDNA5 WGP has 320 KB).
    attn_wmma<<<dim3(MAX_BS * N_HEADS * (SEQ / 16)), blk, ATTN_LDS_BYTES, stream>>>(
        qb, kpanel, vpanel, attn, sp);

    // Output projection.
    gemm_nt_wmma<<<dim3(DIMX / 128, NTOK / 128), blk, GEMM_LDS_BYTES, stream>>>(
        attn, wo, out, NTOK, DIMX, DIMX, 0, sp);
}